// NN_6399501271538
// MI455X (gfx1250) — compile-verified
//
#include <hip/hip_runtime.h>
#include <hip/hip_bf16.h>
#include <math.h>

// ---------------------------------------------------------------------------
// GCN forward: 3x GCNConv (fp32 WMMA GEMM + edge scatter) + mean pool + head.
// Edge scatter dominates bytes; h/agg (25.6MB each) fit in the 192MB L2, so
// the gather/scatter phase runs near L2 bandwidth. GEMMs are tiny, kept in
// fp32 numerics via V_WMMA_F32_16X16X4_F32 (20 static wmma ops).
// ---------------------------------------------------------------------------

typedef __attribute__((ext_vector_type(2))) float v2f;
typedef __attribute__((ext_vector_type(8))) float v8f;

// Only check the builtin on the DEVICE pass; the host pass doesn't see it.
#if defined(__HIP_DEVICE_COMPILE__)
#if !__has_builtin(__builtin_amdgcn_wmma_f32_16x16x4_f32)
#error "gfx1250 f32 WMMA builtin not available on this toolchain"
#endif
#endif

#define HID     64
#define NGRAPHS 64

__device__ __forceinline__ float sigmoidf_(float x) {
    return 1.0f / (1.0f + __expf(-x));
}

// ---------------------------------------------------------------------------
// Zero-fill (grid-stride).
__global__ void zero_kernel(float* __restrict__ p, int n) {
    int i = blockIdx.x * blockDim.x + threadIdx.x;
    int stride = gridDim.x * blockDim.x;
    for (; i < n; i += stride) p[i] = 0.0f;
}

// deg[dst] += ew  (weighted in-degree)
__global__ void deg_kernel(const int* __restrict__ dst, const float* __restrict__ ew,
                           float* __restrict__ deg, int E) {
    int e = blockIdx.x * blockDim.x + threadIdx.x;
    if (e < E) unsafeAtomicAdd(&deg[dst[e]], ew[e]);
}

// dinv = rsqrt(max(deg + 1, eps))   (+1 = self loop)
__global__ void dinv_kernel(const float* __restrict__ deg, float* __restrict__ dinv, int N) {
    int i = blockIdx.x * blockDim.x + threadIdx.x;
    if (i < N) {
        float d = deg[i] + 1.0f;
        dinv[i] = rsqrtf(fmaxf(d, 1e-12f));
    }
}

// ---------------------------------------------------------------------------
// h[N,64] = act(xin[N,K]) @ W[K,64] via V_WMMA_F32_16X16X4_F32.
// One wave -> one 16x16 tile of h. Block = 128 thr = 4 waves = 4 column tiles.
// A frag (16x4 f32): lanes 0-15 hold K=0,1 ; lanes 16-31 hold K=2,3 (row = lane&15).
// B frag (4x16 f32): same K split, col = lane&15.
// C/D (16x16 f32): VGPR v -> row v (lanes 0-15) / row v+8 (lanes 16-31).
template <bool SIG, int K>
__global__ void gemm_wmma(const float* __restrict__ xin, const float* __restrict__ W,
                          float* __restrict__ h, int N) {
    const int lane  = threadIdx.x & 31;
    const int warp  = threadIdx.x >> 5;
    const int m0    = blockIdx.x * 16;
    const int n0    = warp * 16;
    const int r     = lane & 15;          // row-in-tile for A, col-in-tile for B/D
    const int khalf = (lane >> 4) << 1;   // 0 or 2

    int rowA = m0 + r;
    if (rowA > N - 1) rowA = N - 1;       // clamp loads (partial-tile safety)

    v8f c = {};
#pragma unroll
    for (int k = 0; k < K; k += 4) {
        v2f a, b;
        const float* ap = xin + (size_t)rowA * K + (k + khalf);
        a.x = ap[0];
        a.y = ap[1];
        if (SIG) { a.x = sigmoidf_(a.x); a.y = sigmoidf_(a.y); }
        const int kb = k + khalf;
        b.x = W[(size_t)kb * HID + n0 + r];
        b.y = W[(size_t)(kb + 1) * HID + n0 + r];
        c = __builtin_amdgcn_wmma_f32_16x16x4_f32(
                /*neg_a=*/false, a, /*neg_b=*/false, b,
                /*c_mod=*/(short)0, c, /*reuse_a=*/false, /*reuse_b=*/false);
    }

    const int rbase = m0 + ((lane >> 4) << 3);
    const int col   = n0 + r;

    if (m0 + 16 <= N) {
        // Full tile (always taken when N % 16 == 0): wave-uniform branch,
        // straight-line stores, no per-lane exec-mask churn.
#pragma unroll
        for (int v = 0; v < 8; ++v)
            h[(size_t)(rbase + v) * HID + col] = c[v];
    } else {
#pragma unroll
        for (int v = 0; v < 8; ++v) {
            int row = rbase + v;
            if (row < N) h[(size_t)row * HID + col] = c[v];
        }
    }
}

// ---------------------------------------------------------------------------
// Edge scatter: agg[dst] += (dinv[src]*ew*dinv[dst]) * h[src].
// One wave32 per edge; each lane moves float2 -> wave covers 64 floats (256B).
__global__ void scatter_kernel(const float* __restrict__ h,
                               const int* __restrict__ src, const int* __restrict__ dst,
                               const float* __restrict__ ew, const float* __restrict__ dinv,
                               float* __restrict__ agg, int E) {
    int e    = blockIdx.x * (blockDim.x >> 5) + (threadIdx.x >> 5);
    int lane = threadIdx.x & 31;
    if (e >= E) return;
    int s = src[e];
    int d = dst[e];
    float nrm = dinv[s] * ew[e] * dinv[d];
    const float2* hr = (const float2*)(h + (size_t)s * HID);
    float2 v = hr[lane];
    float* ar = agg + (size_t)d * HID + lane * 2;
    unsafeAtomicAdd(ar + 0, v.x * nrm);
    unsafeAtomicAdd(ar + 1, v.y * nrm);
}

// In-place: agg[i,f] += h[i,f] * dinv[i]^2 + b[f]   (self-loop term + bias)
__global__ void selfloop_kernel(float* __restrict__ agg, const float* __restrict__ h,
                                const float* __restrict__ dinv, const float* __restrict__ b,
                                int total) {
    int idx = blockIdx.x * blockDim.x + threadIdx.x;
    if (idx < total) {
        int i = idx >> 6;       // / HID
        int f = idx & 63;       // % HID
        float di = dinv[i];
        agg[idx] = agg[idx] + h[idx] * di * di + b[f];
    }
}

// Mean-pool accumulation: sums[g] += x[i], counts[g] += 1.
__global__ void pool_kernel(const float* __restrict__ x, const int* __restrict__ batch,
                            float* __restrict__ sums, float* __restrict__ cnts, int N) {
    int i = blockIdx.x;
    int f = threadIdx.x;
    if (i >= N) return;
    int g = batch[i];
    unsafeAtomicAdd(&sums[(size_t)g * HID + f], x[(size_t)i * HID + f]);
    if (f == 0) unsafeAtomicAdd(&cnts[g], 1.0f);
}

// out[g] = (sums[g]/max(cnt,1)) . Wout + bout
__global__ void head_kernel(const float* __restrict__ sums, const float* __restrict__ cnts,
                            const float* __restrict__ Wout, const float* __restrict__ bout,
                            float* __restrict__ out) {
    int g = threadIdx.x;
    if (g < NGRAPHS) {
        float cnt = fmaxf(cnts[g], 1.0f);
        float acc = bout[0];
#pragma unroll 8
        for (int f = 0; f < HID; ++f)
            acc += (sums[(size_t)g * HID + f] / cnt) * Wout[f];
        out[g] = acc;
    }
}

// ---------------------------------------------------------------------------
extern "C" void kernel_launch(void* const* d_in, const int* in_sizes, int n_in,
                              void* d_out, int out_size, void* d_ws, size_t ws_size,
                              hipStream_t stream) {
    (void)n_in; (void)out_size; (void)ws_size;

    const float* x     = (const float*)d_in[0];   // [N,16]
    const int*   ei    = (const int*)d_in[1];     // [2,E]
    const float* ew    = (const float*)d_in[2];   // [E]
    const int*   batch = (const int*)d_in[3];     // [N]
    const float* W1    = (const float*)d_in[4];
    const float* b1    = (const float*)d_in[5];
    const float* W2    = (const float*)d_in[6];
    const float* b2    = (const float*)d_in[7];
    const float* W3    = (const float*)d_in[8];
    const float* b3    = (const float*)d_in[9];
    const float* Wout  = (const float*)d_in[10];
    const float* bout  = (const float*)d_in[11];
    float*       out   = (float*)d_out;

    const int N = in_sizes[3];
    const int E = in_sizes[2];

    // Workspace layout (floats)
    float* ws   = (float*)d_ws;
    float* deg  = ws;                 ws += N;
    float* dinv = ws;                 ws += N;
    float* h    = ws;                 ws += (size_t)N * HID;
    float* agg  = ws;                 ws += (size_t)N * HID;
    float* sums = ws;                 ws += NGRAPHS * HID;
    float* cnts = ws;                 /* ws += NGRAPHS; */

    const int* src = ei;
    const int* dst = ei + E;

    const int total  = N * HID;
    const int mtiles = (N + 15) / 16;
    const int zb     = 2048;                       // zero-kernel grid (grid-stride)

    // Degrees (shared by all 3 layers)
    zero_kernel<<<zb, 256, 0, stream>>>(deg, N);
    deg_kernel<<<(E + 255) / 256, 256, 0, stream>>>(dst, ew, deg, E);
    dinv_kernel<<<(N + 255) / 256, 256, 0, stream>>>(deg, dinv, N);

    // ---- Layer 1: h = x @ W1 ; agg = scatter(h) + self-loop + b1 ----
    gemm_wmma<false, 16><<<mtiles, 128, 0, stream>>>(x, W1, h, N);
    zero_kernel<<<zb, 256, 0, stream>>>(agg, total);
    scatter_kernel<<<(E + 7) / 8, 256, 0, stream>>>(h, src, dst, ew, dinv, agg, E);
    selfloop_kernel<<<(total + 255) / 256, 256, 0, stream>>>(agg, h, dinv, b1, total);

    // ---- Layer 2: h = sigmoid(agg) @ W2 ; agg = scatter + self-loop + b2 ----
    gemm_wmma<true, 64><<<mtiles, 128, 0, stream>>>(agg, W2, h, N);
    zero_kernel<<<zb, 256, 0, stream>>>(agg, total);
    scatter_kernel<<<(E + 7) / 8, 256, 0, stream>>>(h, src, dst, ew, dinv, agg, E);
    selfloop_kernel<<<(total + 255) / 256, 256, 0, stream>>>(agg, h, dinv, b2, total);

    // ---- Layer 3 ----
    gemm_wmma<true, 64><<<mtiles, 128, 0, stream>>>(agg, W3, h, N);
    zero_kernel<<<zb, 256, 0, stream>>>(agg, total);
    scatter_kernel<<<(E + 7) / 8, 256, 0, stream>>>(h, src, dst, ew, dinv, agg, E);
    selfloop_kernel<<<(total + 255) / 256, 256, 0, stream>>>(agg, h, dinv, b3, total);

    // ---- Mean pool + head ----
    zero_kernel<<<8, 256, 0, stream>>>(sums, NGRAPHS * HID + NGRAPHS); // sums+cnts contiguous
    pool_kernel<<<N, HID, 0, stream>>>(agg, batch, sums, cnts, N);
    head_kernel<<<1, 64, 0, stream>>>(sums, cnts, Wout, bout, out);
}